// Struc2Vec_86380382257163
// MI455X (gfx1250) — compile-verified
//
#include <hip/hip_runtime.h>
#include <cstdint>
#include <cstddef>

#define PDIM 128
#define WSTRIDE 130  // padded LDS row stride (words): lanes land 2 banks apart

typedef __attribute__((ext_vector_type(2))) float v2f;
typedef __attribute__((ext_vector_type(8))) float v8f;

__device__ __forceinline__ float leaky01(float v) { return v > 0.f ? v : 0.01f * v; }

// --- (1) agg_t[dst[e]][j] += leaky(edge_attr[e]*W3[j] + b3[j]) --------------
// one thread = one edge x 4 features
__global__ void edge_t_kernel(const float* __restrict__ edge_attr,
                              const int* __restrict__ dst,
                              const float* __restrict__ W3,
                              const float* __restrict__ b3,
                              float* __restrict__ agg, int E) {
  long long idx = (long long)blockIdx.x * blockDim.x + threadIdx.x;
  long long total = (long long)E * (PDIM / 4);
  long long stride = (long long)gridDim.x * blockDim.x;
  for (; idx < total; idx += stride) {
    int e = (int)(idx >> 5);
    int j0 = (int)(idx & 31) * 4;
    float a = edge_attr[e];
    float4 w = *(const float4*)(W3 + j0);
    float4 b = *(const float4*)(b3 + j0);
    float* p = agg + (long long)dst[e] * PDIM + j0;
    atomicAdd(p + 0, leaky01(a * w.x + b.x));
    atomicAdd(p + 1, leaky01(a * w.y + b.y));
    atomicAdd(p + 2, leaky01(a * w.z + b.z));
    atomicAdd(p + 3, leaky01(a * w.w + b.w));
  }
}

// --- (2) base[i][j] = type_embed(i)[j] + b1[j] + b2[j] ----------------------
__global__ void type_base_kernel(const float* __restrict__ xv, const float* __restrict__ xp,
                                 const float* __restrict__ xd,
                                 const float* __restrict__ Wv, const float* __restrict__ bv,
                                 const float* __restrict__ Wp, const float* __restrict__ bp,
                                 const float* __restrict__ Wd, const float* __restrict__ bd,
                                 const float* __restrict__ b1, const float* __restrict__ b2,
                                 float* __restrict__ base, int NVn, int NPn, int NDn) {
  long long total = (long long)(NVn + NPn + NDn) * PDIM;
  long long idx = (long long)blockIdx.x * blockDim.x + threadIdx.x;
  long long stride = (long long)gridDim.x * blockDim.x;
  for (; idx < total; idx += stride) {
    int i = (int)(idx >> 7);
    int j = (int)(idx & (PDIM - 1));
    float s;
    if (i < NVn) {
      s = xv[i * 2] * Wv[j] + xv[i * 2 + 1] * Wv[PDIM + j] + bv[j];
    } else if (i < NVn + NPn) {
      int k = i - NVn;
      s = xp[k * 3] * Wp[j] + xp[k * 3 + 1] * Wp[PDIM + j] + xp[k * 3 + 2] * Wp[2 * PDIM + j] + bp[j];
    } else {
      int k = i - NVn - NPn;
      s = xd[k * 2] * Wd[j] + xd[k * 2 + 1] * Wd[PDIM + j] + bd[j];
    }
    base[idx] = s + b1[j] + b2[j];
  }
}

// --- (3) agg[dst[e]][j] += mu[src[e]][j] ; one thread = edge x 4 features ---
__global__ void scatter_mu_kernel(const float* __restrict__ mu,
                                  const int* __restrict__ src,
                                  const int* __restrict__ dst,
                                  float* __restrict__ agg, int E) {
  long long idx = (long long)blockIdx.x * blockDim.x + threadIdx.x;
  long long total = (long long)E * (PDIM / 4);
  long long stride = (long long)gridDim.x * blockDim.x;
  for (; idx < total; idx += stride) {
    int e = (int)(idx >> 5);
    int j0 = (int)(idx & 31) * 4;
    float4 v = *(const float4*)(mu + (long long)src[e] * PDIM + j0);
    float* p = agg + (long long)dst[e] * PDIM + j0;
    atomicAdd(p + 0, v.x);
    atomicAdd(p + 1, v.y);
    atomicAdd(p + 2, v.z);
    atomicAdd(p + 3, v.w);
  }
}

// --- (4) out = act(A @ W + add); W is (128,128) row-major (in,out) ----------
// One wave computes a 16(M)x128(N) strip with V_WMMA_F32_16X16X4_F32.
// W staged to LDS TRANSPOSED with padded stride so each B fragment is one
// aligned ds_load_b64 (no VGPR repacking):  Ws[n*WSTRIDE + k] = W[k*128 + n].
// A 16x4 frag: lanes 0-15 -> M=lane, K=0,1 ; lanes 16-31 -> M=lane-16, K=2,3
// B 4x16 frag: lanes 0-15 -> K=k0,k0+1    ; lanes 16-31 -> K=k0+2,k0+3 ; N=lane&15
// C/D 16x16:  VGPR r -> M=r (lanes 0-15) / M=r+8 (lanes 16-31), N=lane&15
__global__ void __launch_bounds__(128)
gemm_wmma_kernel(const float* __restrict__ A, const float* __restrict__ W,
                 const float* __restrict__ addv, float* __restrict__ outp,
                 int nRows, int applyLeaky) {
  __shared__ float Ws[PDIM * WSTRIDE];  // 66.5 KB (WGP has 320 KB LDS)
  for (int i = threadIdx.x * 4; i < PDIM * PDIM; i += 128 * 4) {
    float4 w = *(const float4*)(W + i);   // coalesced read of row k, cols n..n+3
    int k = i >> 7;
    int n = i & (PDIM - 1);
    Ws[(n + 0) * WSTRIDE + k] = w.x;
    Ws[(n + 1) * WSTRIDE + k] = w.y;
    Ws[(n + 2) * WSTRIDE + k] = w.z;
    Ws[(n + 3) * WSTRIDE + k] = w.w;
  }
  __syncthreads();

  const int lane = threadIdx.x & 31;
  const int waveId = threadIdx.x >> 5;
  const int row0 = (blockIdx.x * 4 + waveId) * 16;
  const int half = lane >> 4;   // 0: lanes 0-15, 1: lanes 16-31
  const int l16 = lane & 15;

  int arow = row0 + l16;
  if (arow >= nRows) arow = nRows - 1;  // clamp; invalid rows never stored
  const float* aptr = A + (size_t)arow * PDIM + 2 * half;

  v8f acc[8] = {};
#pragma unroll 4
  for (int ks = 0; ks < 32; ++ks) {
    v2f a = *(const v2f*)(aptr + ks * 4);
    const int k0 = ks * 4 + 2 * half;
#pragma unroll
    for (int nt = 0; nt < 8; ++nt) {
      const int n = nt * 16 + l16;
      v2f b = *(const v2f*)(&Ws[n * WSTRIDE + k0]);  // single ds_load_b64
      acc[nt] = __builtin_amdgcn_wmma_f32_16x16x4_f32(
          false, a, false, b, (short)0, acc[nt], false, false);
    }
  }

#pragma unroll
  for (int nt = 0; nt < 8; ++nt) {
#pragma unroll
    for (int r = 0; r < 8; ++r) {
      int row = row0 + r + 8 * half;
      if (row < nRows) {
        int col = nt * 16 + l16;
        size_t o = (size_t)row * PDIM + col;
        float v = acc[nt][r] + addv[o];
        if (applyLeaky) v = leaky01(v);
        outp[o] = v;
      }
    }
  }
}

// --- (5) graph pooling ------------------------------------------------------
__global__ void pool_sum_kernel(const float* __restrict__ mu,
                                const int* __restrict__ batch,
                                float* __restrict__ gsum, int N) {
  long long idx = (long long)blockIdx.x * blockDim.x + threadIdx.x;
  long long total = (long long)N * (PDIM / 4);
  long long stride = (long long)gridDim.x * blockDim.x;
  for (; idx < total; idx += stride) {
    int i = (int)(idx >> 5);
    int j0 = (int)(idx & 31) * 4;
    float4 v = *(const float4*)(mu + (long long)i * PDIM + j0);
    float* p = gsum + (long long)batch[i] * PDIM + j0;
    atomicAdd(p + 0, v.x);
    atomicAdd(p + 1, v.y);
    atomicAdd(p + 2, v.z);
    atomicAdd(p + 3, v.w);
  }
}

__global__ void pool_cnt_kernel(const int* __restrict__ batch,
                                float* __restrict__ gcnt, int N) {
  int i = blockIdx.x * blockDim.x + threadIdx.x;
  if (i < N) atomicAdd(&gcnt[batch[i]], 1.0f);
}

__global__ void readout_kernel(const float* __restrict__ gsum,
                               const float* __restrict__ gcnt,
                               const float* __restrict__ Wc,
                               const float* __restrict__ bc,
                               float* __restrict__ out, int G) {
  int g = threadIdx.x + blockIdx.x * blockDim.x;
  if (g >= G) return;
  float c = gcnt[g];
  if (c < 1.f) c = 1.f;
  float s = bc[0];
  for (int j = 0; j < PDIM; ++j) s += (gsum[g * PDIM + j] / c) * Wc[j];
  out[g] = 1.f / (1.f + expf(-s));
}

extern "C" void kernel_launch(void* const* d_in, const int* in_sizes, int n_in,
                              void* d_out, int out_size, void* d_ws, size_t ws_size,
                              hipStream_t stream) {
  const float* edge_attr = (const float*)d_in[0];
  const float* node_mu   = (const float*)d_in[1];
  const float* xv = (const float*)d_in[2];
  const float* xp = (const float*)d_in[3];
  const float* xd = (const float*)d_in[4];
  const float* W1 = (const float*)d_in[5];  const float* b1 = (const float*)d_in[6];
  const float* W2 = (const float*)d_in[7];  const float* b2 = (const float*)d_in[8];
  const float* W3 = (const float*)d_in[9];  const float* b3 = (const float*)d_in[10];
  const float* Wv = (const float*)d_in[11]; const float* bv = (const float*)d_in[12];
  const float* Wp = (const float*)d_in[13]; const float* bp = (const float*)d_in[14];
  const float* Wd = (const float*)d_in[15]; const float* bd = (const float*)d_in[16];
  const float* Wc = (const float*)d_in[17]; const float* bc = (const float*)d_in[18];
  const int* edge_index = (const int*)d_in[19];
  const int* batch = (const int*)d_in[21];

  const int E = in_sizes[0];
  const int N = in_sizes[1] / PDIM;
  const int NVn = in_sizes[2] / 2;
  const int NPn = in_sizes[3] / 3;
  const int NDn = in_sizes[4] / 2;
  const int G = out_size;  // output is (G,1)
  const int* srcIdx = edge_index;
  const int* dstIdx = edge_index + E;

  // Workspace layout (all f32): base | bufA | bufB | gsum | gcnt
  float* base = (float*)d_ws;
  float* bufA = base + (size_t)N * PDIM;
  float* bufB = bufA + (size_t)N * PDIM;
  float* gsum = bufB + (size_t)N * PDIM;
  float* gcnt = gsum + (size_t)G * PDIM;
  const size_t featBytes = (size_t)N * PDIM * sizeof(float);

  const long long tE4 = (long long)E * (PDIM / 4);
  const int bE = (int)((tE4 + 255) / 256);
  const long long tN = (long long)N * PDIM;
  const int bN = (int)((tN + 255) / 256);
  const long long tN4 = (long long)N * (PDIM / 4);
  const int bN4 = (int)((tN4 + 255) / 256);
  const int mTiles = (N + 15) / 16;
  const int gBlocks = (mTiles + 3) / 4;  // 4 waves (M-tiles) per block

  // agg_t = segment_sum(leaky(edge_attr*W3+b3), dst)   (into bufA)
  hipMemsetAsync(bufA, 0, featBytes, stream);
  edge_t_kernel<<<bE, 256, 0, stream>>>(edge_attr, dstIdx, W3, b3, bufA, E);

  // base = type_add + b1 + b2
  type_base_kernel<<<bN, 256, 0, stream>>>(xv, xp, xd, Wv, bv, Wp, bp, Wd, bd,
                                           b1, b2, base, NVn, NPn, NDn);

  // base += agg_t @ W2
  gemm_wmma_kernel<<<gBlocks, 128, 0, stream>>>(bufA, W2, base, base, N, 0);

  // 4 message-passing iterations: agg = scatter(mu), mu = leaky(agg@W1 + base)
  const float* cur = node_mu;
  float* bufs[2] = {bufB, bufA};
  for (int it = 0; it < 4; ++it) {
    float* dst = bufs[it & 1];
    hipMemsetAsync(dst, 0, featBytes, stream);
    scatter_mu_kernel<<<bE, 256, 0, stream>>>(cur, srcIdx, dstIdx, dst, E);
    gemm_wmma_kernel<<<gBlocks, 128, 0, stream>>>(dst, W1, base, dst, N, 1);
    cur = dst;
  }

  // graph mean + sigmoid readout
  hipMemsetAsync(gsum, 0, (size_t)G * PDIM * sizeof(float), stream);
  hipMemsetAsync(gcnt, 0, (size_t)G * sizeof(float), stream);
  pool_sum_kernel<<<bN4, 256, 0, stream>>>(cur, batch, gsum, N);
  pool_cnt_kernel<<<(N + 255) / 256, 256, 0, stream>>>(batch, gcnt, N);
  readout_kernel<<<(G + 63) / 64, 64, 0, stream>>>(gsum, gcnt, Wc, bc, (float*)d_out, G);
}